// Critic_76965813944960
// MI455X (gfx1250) — compile-verified
//
#include <hip/hip_runtime.h>
#include <hip/hip_bf16.h>

#define N_NODES 10000
#define BATCH   32
#define F_IN    5
#define HID     128
#define LN_EPS  1e-5f

typedef float v2f __attribute__((ext_vector_type(2)));
typedef float v8f __attribute__((ext_vector_type(8)));

// GCN symmetric-norm coefficients for the bidirectional chain + self loops.
// deg = 2 at endpoints, 3 interior -> only three distinct constants.
#define RS9 0.3333333432674408f   // rsqrt(9) == 1/3 rounded to f32
#define RS6 0.4082482904638631f   // rsqrt(6)

// Node n's 5 features are contiguous within exactly one of state/action:
// concat boundary (40000 floats) falls precisely at node 8000.
__device__ __forceinline__ const float* node_ptr(const float* __restrict__ state,
                                                 const float* __restrict__ action,
                                                 int b, int n) {
    return (n < (N_NODES * 4) / F_IN)
               ? state + b * (N_NODES * 4) + n * F_IN
               : action + b * N_NODES + n * F_IN - N_NODES * 4;
}

__global__ void zero_pool_kernel(float* __restrict__ pooled) {
    pooled[threadIdx.x] = 0.0f;
}

__global__ void __launch_bounds__(256)
gcn_ln_pool_kernel(const float* __restrict__ state,
                   const float* __restrict__ action,
                   const float* __restrict__ conv_w,   // [F_IN, HID] row-major
                   const float* __restrict__ conv_b,   // [HID]
                   const float* __restrict__ ln_g,
                   const float* __restrict__ ln_b,
                   float* __restrict__ pooled,         // [HID] global accumulator
                   int n_tasks, int n_waves_total)
{
    __shared__ float lds_pool[HID];
    const int tid  = threadIdx.x;
    const int lane = tid & 31;
    const int wave = tid >> 5;
    const int m16  = lane & 15;
    const bool lo  = lane < 16;

    if (tid < HID) lds_pool[tid] = 0.0f;
    __syncthreads();

    // Preload B-matrix tiles of conv_w (K=0..3 pair + K=4 pad row), bias, LN params.
    // B 4x16 layout: lane = column N, VGPR0/1 = K; lanes 0-15 hold K0/K1, 16-31 hold K2/K3.
    v2f   bk[8];
    float bk4[8], cb[8], lg[8], lb[8];
#pragma unroll
    for (int j = 0; j < 8; ++j) {
        const int col = m16 + 16 * j;
        bk[j].x = conv_w[(lo ? 0 : 2) * HID + col];
        bk[j].y = conv_w[(lo ? 1 : 3) * HID + col];
        bk4[j]  = lo ? conv_w[4 * HID + col] : 0.0f;
        cb[j]   = conv_b[col];
        lg[j]   = ln_g[col];
        lb[j]   = ln_b[col];
    }

    float pool[8];
#pragma unroll
    for (int j = 0; j < 8; ++j) pool[j] = 0.0f;

    const int gwave = blockIdx.x * (blockDim.x >> 5) + wave;
    for (int task = gwave; task < n_tasks; task += n_waves_total) {
        const int b    = task / (N_NODES / 16);
        const int tile = task - b * (N_NODES / 16);
        const int gn   = tile * 16 + m16;   // node handled by this lane (A-matrix row)

        // Stencil coefficients as constant selects (no runtime division/rsqrt).
        const bool end0 = (gn == 0), endN = (gn == N_NODES - 1);
        const float a_self = (end0 || endN) ? 0.5f : RS9;
        const float a_prev = end0 ? 0.0f : ((gn == 1 || endN) ? RS6 : RS9);
        const float a_next = endN ? 0.0f : ((end0 || gn == N_NODES - 2) ? RS6 : RS9);
        const int   np = end0 ? gn : gn - 1;
        const int   nn = endN ? gn : gn + 1;

        const float* __restrict__ p_prev = node_ptr(state, action, b, np);
        const float* __restrict__ p_self = node_ptr(state, action, b, gn);
        const float* __restrict__ p_next = node_ptr(state, action, b, nn);

        float v[F_IN];
#pragma unroll
        for (int f = 0; f < F_IN; ++f)
            v[f] = a_self * p_self[f] + a_prev * p_prev[f] + a_next * p_next[f];

        // A 16x4 layout: lane = row M (mod 16); lanes 0-15 carry K0/K1, 16-31 carry K2/K3.
        v2f a1, a2;
        a1.x = lo ? v[0] : v[2];
        a1.y = lo ? v[1] : v[3];
        a2.x = lo ? v[4] : 0.0f;   // K=4 in padded second WMMA
        a2.y = 0.0f;

        v8f c[8];
#pragma unroll
        for (int j = 0; j < 8; ++j) {
            v8f z = {};
            c[j] = __builtin_amdgcn_wmma_f32_16x16x4_f32(
                false, a1, false, bk[j], (short)0, z, false, false);
            v2f b2v; b2v.x = bk4[j]; b2v.y = 0.0f;
            c[j] = __builtin_amdgcn_wmma_f32_16x16x4_f32(
                false, a2, false, b2v, (short)0, c[j], false, false);
        }

        // bias + ReLU (D layout: VGPR r = rows M=r (lanes 0-15) / M=r+8 (lanes 16-31))
#pragma unroll
        for (int j = 0; j < 8; ++j) {
#pragma unroll
            for (int r = 0; r < 8; ++r) {
                const float x = c[j][r] + cb[j];
                c[j][r] = x > 0.0f ? x : 0.0f;
            }
        }

        // LayerNorm per node row, then accumulate pooled partial sums.
#pragma unroll
        for (int r = 0; r < 8; ++r) {
            float s = 0.0f, q = 0.0f;
#pragma unroll
            for (int j = 0; j < 8; ++j) { const float x = c[j][r]; s += x; q += x * x; }
#pragma unroll
            for (int off = 8; off >= 1; off >>= 1) {   // width-16: halves stay independent
                s += __shfl_xor(s, off, 16);
                q += __shfl_xor(q, off, 16);
            }
            const float mean = s * (1.0f / HID);
            const float var  = q * (1.0f / HID) - mean * mean;
            const float inv  = rsqrtf(var + LN_EPS);
#pragma unroll
            for (int j = 0; j < 8; ++j)
                pool[j] += (c[j][r] - mean) * inv * lg[j] + lb[j];
        }
    }

    // Fold lane halves (lane l and l+16 hold the same channel, different rows).
#pragma unroll
    for (int j = 0; j < 8; ++j) {
        const float t = pool[j] + __shfl_xor(pool[j], 16, 32);
        if (lo) atomicAdd(&lds_pool[16 * j + m16], t);
    }
    __syncthreads();
    if (tid < HID) atomicAdd(&pooled[tid], lds_pool[tid]);
}

__global__ void __launch_bounds__(128)
mlp_head_kernel(const float* __restrict__ pooled,
                const float* __restrict__ w2, const float* __restrict__ b2,
                const float* __restrict__ w3, const float* __restrict__ b3,
                float* __restrict__ out)
{
    __shared__ float sp[HID];
    __shared__ float red[HID];
    const int h = threadIdx.x;
    sp[h] = pooled[h];
    __syncthreads();
    float acc = b2[h];
#pragma unroll 8
    for (int k = 0; k < HID; ++k) acc = fmaf(sp[k], w2[k * HID + h], acc);
    const float y = acc > 0.0f ? acc : 0.0f;
    red[h] = y * w3[h];
    __syncthreads();
    for (int s = 64; s >= 1; s >>= 1) {
        if (h < s) red[h] += red[h + s];
        __syncthreads();
    }
    if (h == 0) out[0] = red[0] + b3[0];
}

extern "C" void kernel_launch(void* const* d_in, const int* in_sizes, int n_in,
                              void* d_out, int out_size, void* d_ws, size_t ws_size,
                              hipStream_t stream) {
    (void)in_sizes; (void)n_in; (void)out_size; (void)ws_size;
    const float* state  = (const float*)d_in[0];
    const float* action = (const float*)d_in[1];
    const float* conv_w = (const float*)d_in[2];
    const float* conv_b = (const float*)d_in[3];
    const float* ln_g   = (const float*)d_in[4];
    const float* ln_b   = (const float*)d_in[5];
    const float* w2     = (const float*)d_in[6];
    const float* b2     = (const float*)d_in[7];
    const float* w3     = (const float*)d_in[8];
    const float* b3     = (const float*)d_in[9];
    float* pooled = (float*)d_ws;
    float* out    = (float*)d_out;

    zero_pool_kernel<<<1, HID, 0, stream>>>(pooled);

    const int n_tasks = BATCH * (N_NODES / 16);   // 20000 node-tiles
    const int blocks  = 250;                      // 2000 waves -> 10 tiles/wave
    gcn_ln_pool_kernel<<<blocks, 256, 0, stream>>>(
        state, action, conv_w, conv_b, ln_g, ln_b, pooled,
        n_tasks, blocks * 8);

    mlp_head_kernel<<<1, HID, 0, stream>>>(pooled, w2, b2, w3, b3, out);
}